// VariationalDynamicLatentVariable_44435731644633
// MI455X (gfx1250) — compile-verified
//
#include <hip/hip_runtime.h>
#include <math.h>

// N=2000, D=8, H=16, HD=128
// cov_n = kron(M_n, I_16)  with  M_n = s_n*(t t^T + c*J) + 1e-6*I  (8x8)
// => L_n = kron(chol8(M_n), I_16);  out = mu + kron-structured matmul
// Per row: D(16x16) = E^T(16x8) * L^T(8x16-padded) + MU   via two f32 WMMAs (K=4 each).

typedef __attribute__((ext_vector_type(2))) float v2f;
typedef __attribute__((ext_vector_type(4))) float v4f;
typedef __attribute__((ext_vector_type(8))) float v8f;

#define HD 128

__device__ __forceinline__ float sigmoidf_(float x) {
    return 1.0f / (1.0f + __expf(-x));
}

__global__ __launch_bounds__(256)
void vdlv_rsample_kernel(const float* __restrict__ q_mu,
                         const float* __restrict__ q_dim_v,
                         const float* __restrict__ q_dim_B,
                         const float* __restrict__ q_time_var,
                         const float* __restrict__ eps,
                         float* __restrict__ out,
                         int N)
{
    __shared__ float lchol[8][64];            // per-wave 8x8 L staging (8 waves/block)

    const int lane = threadIdx.x & 31;
    const int wv   = threadIdx.x >> 5;
    const int n    = blockIdx.x * 8 + wv;     // one row per wave (wave-uniform -> full EXEC)
    if (n >= N) return;

    // ---- wave-uniform small parameters -------------------------------------
    float t[8];
#pragma unroll
    for (int i = 0; i < 8; ++i)
        t[i] = 2.0f * sigmoidf_(q_dim_B[i]) - 1.0f;
    const float c = sigmoidf_(q_dim_v[0]);
    const float s = sigmoidf_(q_time_var[n]);

    // ---- M = s*(t t^T + c) + 1e-6*I  (lower), zeros above diagonal ---------
    float Lm[8][8];
#pragma unroll
    for (int i = 0; i < 8; ++i)
#pragma unroll
        for (int j = 0; j < 8; ++j)
            Lm[i][j] = (j <= i)
                     ? (s * (t[i] * t[j] + c) + ((i == j) ? 1e-6f : 0.0f))
                     : 0.0f;

    // ---- unrolled in-place 8x8 lower Cholesky (all indices compile-time) ---
#pragma unroll
    for (int k = 0; k < 8; ++k) {
        Lm[k][k] = sqrtf(Lm[k][k]);
        const float inv = 1.0f / Lm[k][k];
#pragma unroll
        for (int i = k + 1; i < 8; ++i) Lm[i][k] *= inv;
#pragma unroll
        for (int j = k + 1; j < 8; ++j)
#pragma unroll
            for (int i = j; i < 8; ++i) Lm[i][j] -= Lm[i][k] * Lm[j][k];
    }

    // ---- stage L into this wave's LDS slice (lane 0; values wave-uniform) --
    if (lane == 0) {
#pragma unroll
        for (int i = 0; i < 8; ++i) {
            v4f lo = { Lm[i][0], Lm[i][1], Lm[i][2], Lm[i][3] };
            v4f hi = { Lm[i][4], Lm[i][5], Lm[i][6], Lm[i][7] };
            *reinterpret_cast<v4f*>(&lchol[wv][i * 8 + 0]) = lo;
            *reinterpret_cast<v4f*>(&lchol[wv][i * 8 + 4]) = hi;
        }
    }
    // DS ops are in-order within a wave; fence the compiler + wait DScnt.
    asm volatile("s_wait_dscnt 0" ::: "memory");

    // ---- build B (L^T, columns j>=8 zero) ----------------------------------
    const int ncol = lane & 15;               // output column j (latent dim)
    const int hi16 = lane >> 4;               // 0: K=0,1 / 1: K=2,3 (per chunk)
    const int rsel = ncol & 7;
    v2f b0 = *reinterpret_cast<const v2f*>(&lchol[wv][rsel * 8 + hi16 * 2]);
    v2f b1 = *reinterpret_cast<const v2f*>(&lchol[wv][rsel * 8 + hi16 * 2 + 4]);
    const float vmask = (ncol < 8) ? 1.0f : 0.0f;
    b0 *= vmask;
    b1 *= vmask;

    // ---- A = E^T chunks (K=0..3 and K=4..7), per ISA 16x4 f32 layout -------
    const float* erow = eps + (size_t)n * HD;
    const int t0 = (lane < 16) ? lane : (lane + 16);
    v2f a0 = { erow[t0],      erow[t0 + 16] };
    v2f a1 = { erow[t0 + 64], erow[t0 + 80] };

    // ---- C = q_mu in D layout: 8 contiguous floats per lane ----------------
    const size_t obase = (size_t)n * HD + (size_t)(rsel * 16 + hi16 * 8);
    v4f c_lo = *reinterpret_cast<const v4f*>(q_mu + obase);
    v4f c_hi = *reinterpret_cast<const v4f*>(q_mu + obase + 4);
    v8f acc = { c_lo.x, c_lo.y, c_lo.z, c_lo.w, c_hi.x, c_hi.y, c_hi.z, c_hi.w };

    // ---- D = A1*B1 + (A0*B0 + C): two chained f32 WMMAs, full EXEC ---------
    acc = __builtin_amdgcn_wmma_f32_16x16x4_f32(false, a0, false, b0,
                                                (short)0, acc, false, false);
    acc = __builtin_amdgcn_wmma_f32_16x16x4_f32(false, a1, false, b1,
                                                (short)0, acc, false, false);

    // ---- store valid columns (lanes with j<8): two b128 per lane -----------
    if (ncol < 8) {
        float* orow = out + obase;
        v4f o0 = { acc[0], acc[1], acc[2], acc[3] };
        v4f o1 = { acc[4], acc[5], acc[6], acc[7] };
        *reinterpret_cast<v4f*>(orow)     = o0;
        *reinterpret_cast<v4f*>(orow + 4) = o1;
    }
}

extern "C" void kernel_launch(void* const* d_in, const int* in_sizes, int n_in,
                              void* d_out, int out_size, void* d_ws, size_t ws_size,
                              hipStream_t stream) {
    const float* q_mu       = (const float*)d_in[0];
    const float* q_dim_v    = (const float*)d_in[1];
    const float* q_dim_B    = (const float*)d_in[2];
    const float* q_time_var = (const float*)d_in[3];
    const float* eps        = (const float*)d_in[4];
    float* out = (float*)d_out;

    const int N = in_sizes[3];                 // q_time_var has N elements
    const int blocks = (N + 7) / 8;            // 8 waves (rows) per 256-thread block

    hipLaunchKernelGGL(vdlv_rsample_kernel, dim3(blocks), dim3(256), 0, stream,
                       q_mu, q_dim_v, q_dim_B, q_time_var, eps, out, N);
}